// HSM_62508954026539
// MI455X (gfx1250) — compile-verified
//
#include <hip/hip_runtime.h>
#include <hip/hip_bf16.h>
#include <math.h>

// V = 50257, N_DEC = 50256, D = 1024, N_EX = 8192, MAX_LEN = 24
#define HSM_D        1024
#define HSM_MAXLEN   24
#define HSM_TILE     16          // examples per block
#define HSM_WAVES    4           // waves per block
#define HSM_LVL_PW   (HSM_MAXLEN / HSM_WAVES)   // 6 levels per wave
#define HSM_KSTEPS   (HSM_D / 4)                // 256 wmma steps of K=4

typedef float v2f __attribute__((ext_vector_type(2)));
typedef float v8f __attribute__((ext_vector_type(8)));

__global__ __launch_bounds__(HSM_TILE * 8, 4)
void HSM_62508954026539_kernel(const float* __restrict__ x,
                               const float* __restrict__ W,
                               const int*   __restrict__ t,
                               const int*   __restrict__ paths,
                               const float* __restrict__ codes,
                               const int*   __restrict__ lens,
                               float* __restrict__ out)
{
    // x tile in LDS: 16 rows x 1024 f32 = 64 KB, swizzled at float4 (quad)
    // granularity: physical quad = (quad + row) & 255. Keeps each async b128
    // contiguous in LDS, and the ds_load_b64 A-fetch is 64-bank conflict free
    // (lower lanes hit even bank-pairs 2(s+r), upper lanes odd 2(s+r)+1).
    __shared__ float xs[HSM_TILE * HSM_D];

    const int tid      = threadIdx.x;           // 0..127
    const int wave     = tid >> 5;              // 0..3
    const int lane     = tid & 31;              // wave32
    const int exl      = lane & 15;             // example within tile
    const int hi       = lane >> 4;             // 0: K=0..1, 1: K=2..3 halves
    const int tileBase = blockIdx.x * HSM_TILE;

    // ---- async global->LDS stage of the x tile (ASYNCcnt-tracked, no VGPR
    //      data): 32 x global_load_async_to_lds_b128 per wave ----
    {
        const unsigned ldsBase = (unsigned)(uintptr_t)&xs[0];
        const char*    xgb     = (const char*)(x + (size_t)tileBase * HSM_D);
        #pragma unroll
        for (int i = 0; i < (HSM_TILE * HSM_D / 4) / 128; ++i) {
            int quad = tid + 128 * i;           // 0..4095 (row-major float4s)
            int r    = quad >> 8;               // row 0..15
            int qin  = quad & 255;              // quad within row
            unsigned ldsAddr = ldsBase + (r << 12) + (((qin + r) & 255) << 4);
            unsigned long long gaddr =
                (unsigned long long)(uintptr_t)(xgb + ((size_t)quad << 4));
            asm volatile("global_load_async_to_lds_b128 %0, %1, off"
                         :: "v"(ldsAddr), "v"(gaddr) : "memory");
        }
    }

    // per-example path metadata overlaps the async copy
    const int tj   = t[tileBase + exl];
    const int lenj = lens[tj];

    asm volatile("s_wait_asynccnt 0" ::: "memory");
    __syncthreads();

    float acc = 0.0f;

    #pragma unroll 1
    for (int li = 0; li < HSM_LVL_PW; ++li) {
        const int   l    = wave * HSM_LVL_PW + li;
        const int   node = paths[tj * HSM_MAXLEN + l];
        const float cd   = codes[tj * HSM_MAXLEN + l];

        // B gather: lane (exl,hi) streams floats [4s + 2*hi, +1] of W[node]
        const float2* wr2 = (const float2*)(W + (size_t)node * HSM_D) + hi;

        v8f c = {0.f, 0.f, 0.f, 0.f, 0.f, 0.f, 0.f, 0.f};

        #pragma unroll 8
        for (int s = 0; s < HSM_KSTEPS; ++s) {
            // B: 4x16 f32, lane-symmetric to A layout
            float2 bv = wr2[2 * s];
            // A: 16x4 f32 from quad-swizzled LDS
            int q = (s + exl) & 255;
            float2 av = *(const float2*)
                ((const char*)xs + (exl << 12) + (q << 4) + (hi << 3));
            v2f a = {av.x, av.y};
            v2f b = {bv.x, bv.y};
            // D(16x16,f32) += A(16x4) * B(4x16), exact f32
            c = __builtin_amdgcn_wmma_f32_16x16x4_f32(
                    false, a, false, b, (short)0, c, false, false);
        }

        // ---- diagonal extraction: D[j][j] lives at component j&7,
        //      source lane j (j<8) or j+16 (j>=8) ----
        const int srcLane = (exl < 8) ? exl : (exl + 16);
        const int comp    = exl & 7;
        float d = 0.0f;
        #pragma unroll
        for (int v = 0; v < 8; ++v) {
            float tv = __shfl(c[v], srcLane, 32);
            d = (comp == v) ? tv : d;
        }

        // masked stable softplus(-code * dot); branchless so EXEC stays full
        float z    = -cd * d;
        float loss = fmaxf(z, 0.0f) + log1pf(__expf(-fabsf(z)));
        acc += ((hi == 0) && (l < lenj)) ? loss : 0.0f;
    }

    // wave reduction (upper-half lanes contributed 0) + one atomic per wave
    #pragma unroll
    for (int off = 16; off > 0; off >>= 1)
        acc += __shfl_xor(acc, off, 32);
    if (lane == 0)
        atomicAdd(out, acc);
}

extern "C" void kernel_launch(void* const* d_in, const int* in_sizes, int n_in,
                              void* d_out, int out_size, void* d_ws, size_t ws_size,
                              hipStream_t stream) {
    (void)n_in; (void)out_size; (void)d_ws; (void)ws_size;
    const float* x     = (const float*)d_in[0];
    const float* W     = (const float*)d_in[1];
    const int*   t     = (const int*)  d_in[2];
    const int*   paths = (const int*)  d_in[3];
    const float* codes = (const float*)d_in[4];
    const int*   lens  = (const int*)  d_in[5];
    float* out = (float*)d_out;

    const int nEx    = in_sizes[2];             // 8192
    const int blocks = nEx / HSM_TILE;          // 512

    hipMemsetAsync(out, 0, sizeof(float), stream);
    HSM_62508954026539_kernel<<<blocks, HSM_TILE * 8, 0, stream>>>(
        x, W, t, paths, codes, lens, out);
}